// Loss_20736102105562
// MI455X (gfx1250) — compile-verified
//
#include <hip/hip_runtime.h>
#include <hip/hip_bf16.h>
#include <math.h>

#define BS 8
#define NUM_ROT 12
#define N_PTS 1300
#define N_TGT 1300
#define IMG_H 480
#define IMG_W 640
#define N_SEL 1024
#define N_TILES 82          // ceil(1300/16); even
#define N_PAD (N_TILES * 16)

typedef float v2f __attribute__((ext_vector_type(2)));
typedef float v8f __attribute__((ext_vector_type(8)));

// ---------------------------------------------------------------------------
// Kernel 1: per (b, r) pair, compute dmin[b,r] = mean_n min_m || R p_n - t_m ||
// using V_WMMA_F32_16X16X4_F32 with the squared distance folded into the GEMM:
//   A row  = (-2px, -2py, -2pz, 1)
//   B col  = ( tx,   ty,   tz,  |t|^2 )
//   D      = |t|^2 - 2 p.t        (add |p|^2 after the min; monotone-safe)
// B fragments are pre-swizzled in LDS (entry col*2+half is the float2 a lane
// feeds straight into the WMMA B operand). The hot loop is software-pipelined:
// fetch tiles ct+2/ct+3 while the WMMAs + v_min3_num_f32 chain consume ct/ct+1,
// so the ds_load latency is covered instead of a s_wait_dscnt 0 stall.
// ---------------------------------------------------------------------------
__global__ __launch_bounds__(256)
void rot_dmin_kernel(const float* __restrict__ pred_r,
                     const float* __restrict__ model_points,
                     const float* __restrict__ target_r,
                     float* __restrict__ dmin_out)
{
    __shared__ float2 sB[N_PAD * 2];   // B fragments, pre-swizzled per half
    __shared__ float  sRed[8];

    const int b   = blockIdx.x / NUM_ROT;
    const int r   = blockIdx.x % NUM_ROT;
    const int tid = threadIdx.x;

    // quaternion -> 3x3 "m" (pre-transpose matrix of the reference);
    // pred_pts[n, e] = dot(m[e, :], model[n, :])
    const float q0 = pred_r[(b * NUM_ROT + r) * 4 + 0];
    const float q1 = pred_r[(b * NUM_ROT + r) * 4 + 1];
    const float q2 = pred_r[(b * NUM_ROT + r) * 4 + 2];
    const float q3 = pred_r[(b * NUM_ROT + r) * 4 + 3];
    const float m00 = 1.f - 2.f * (q2 * q2 + q3 * q3);
    const float m01 = 2.f * q1 * q2 - 2.f * q0 * q3;
    const float m02 = 2.f * q0 * q2 + 2.f * q1 * q3;
    const float m10 = 2.f * q1 * q2 + 2.f * q3 * q0;
    const float m11 = 1.f - 2.f * (q1 * q1 + q3 * q3);
    const float m12 = -2.f * q0 * q1 + 2.f * q2 * q3;
    const float m20 = -2.f * q0 * q2 + 2.f * q1 * q3;
    const float m21 = 2.f * q0 * q1 + 2.f * q2 * q3;
    const float m22 = 1.f - 2.f * (q1 * q1 + q2 * q2);

    // Stage targets into LDS in B-fragment order; pad with huge |t|^2 sentinel
    // so padded columns never win the min.
    for (int i = tid; i < N_PAD; i += 256) {
        float tx = 0.f, ty = 0.f, tz = 0.f, t2 = 1.0e30f;
        if (i < N_TGT) {
            tx = target_r[(b * N_TGT + i) * 3 + 0];
            ty = target_r[(b * N_TGT + i) * 3 + 1];
            tz = target_r[(b * N_TGT + i) * 3 + 2];
            t2 = tx * tx + ty * ty + tz * tz;
        }
        sB[i * 2 + 0] = make_float2(tx, ty);   // half 0: (K0, K1)
        sB[i * 2 + 1] = make_float2(tz, t2);   // half 1: (K2, K3)
    }
    __syncthreads();

    const int wave = tid >> 5;
    const int lane = tid & 31;
    const int h    = lane >> 4;     // half-wave selects K-pair (0: K0,K1  1: K2,K3)
    const int l    = lane & 15;     // row (for A) / column (for B) index
    const int bofs = l * 2 + h;     // lane's element slot within one column tile

    float rowSum = 0.f;

    for (int rt = wave; rt < N_TILES; rt += 8) {
        const int n  = rt * 16 + l;
        const int nc = (n < N_PTS) ? n : (N_PTS - 1);   // clamp padded rows (masked later)
        const float mx = model_points[(b * N_PTS + nc) * 3 + 0];
        const float my = model_points[(b * N_PTS + nc) * 3 + 1];
        const float mz = model_points[(b * N_PTS + nc) * 3 + 2];
        const float px = m00 * mx + m01 * my + m02 * mz;
        const float py = m10 * mx + m11 * my + m12 * mz;
        const float pz = m20 * mx + m21 * my + m22 * mz;
        const float a2 = px * px + py * py + pz * pz;

        // A fragment (16x4, f32): lane half holds (K0,K1) or (K2,K3) of row l
        v2f A;
        A.x = h ? (-2.f * pz) : (-2.f * px);
        A.y = h ? 1.f         : (-2.f * py);

        v8f minv;
        #pragma unroll
        for (int i = 0; i < 8; ++i) minv[i] = 3.0e38f;

        // ---- software-pipelined hot loop (double-buffered B fragments) ----
        float2 t0 = sB[0 * 32 + bofs];
        float2 t1 = sB[1 * 32 + bofs];
        for (int ct = 0; ct < N_TILES - 2; ct += 2) {
            const float2 u0 = sB[(ct + 2) * 32 + bofs];   // prefetch next pair
            const float2 u1 = sB[(ct + 3) * 32 + bofs];
            v2f B0; B0.x = t0.x; B0.y = t0.y;
            v2f B1; B1.x = t1.x; B1.y = t1.y;
            v8f c = {0.f, 0.f, 0.f, 0.f, 0.f, 0.f, 0.f, 0.f};
            v8f dA = __builtin_amdgcn_wmma_f32_16x16x4_f32(
                false, A, false, B0, (short)0, c, false, false);
            v8f dB = __builtin_amdgcn_wmma_f32_16x16x4_f32(
                false, A, false, B1, (short)0, c, false, false);
            #pragma unroll
            for (int i = 0; i < 8; ++i) {
                float mi = minv[i];
                asm("v_min3_num_f32 %0, %1, %2, %3"
                    : "=v"(mi) : "v"(mi), "v"(dA[i]), "v"(dB[i]));
                minv[i] = mi;
            }
            t0 = u0; t1 = u1;
        }
        {   // epilogue: last tile pair (no further prefetch)
            v2f B0; B0.x = t0.x; B0.y = t0.y;
            v2f B1; B1.x = t1.x; B1.y = t1.y;
            v8f c = {0.f, 0.f, 0.f, 0.f, 0.f, 0.f, 0.f, 0.f};
            v8f dA = __builtin_amdgcn_wmma_f32_16x16x4_f32(
                false, A, false, B0, (short)0, c, false, false);
            v8f dB = __builtin_amdgcn_wmma_f32_16x16x4_f32(
                false, A, false, B1, (short)0, c, false, false);
            #pragma unroll
            for (int i = 0; i < 8; ++i) {
                float mi = minv[i];
                asm("v_min3_num_f32 %0, %1, %2, %3"
                    : "=v"(mi) : "v"(mi), "v"(dA[i]), "v"(dB[i]));
                minv[i] = mi;
            }
        }

        // Butterfly min across the 16 lanes of each half (rows live per-VGPR).
        #pragma unroll
        for (int mask = 1; mask < 16; mask <<= 1) {
            #pragma unroll
            for (int i = 0; i < 8; ++i) {
                const float o = __shfl_xor(minv[i], mask, 32);
                float mi = minv[i];
                asm("v_min_num_f32 %0, %1, %2" : "=v"(mi) : "v"(mi), "v"(o));
                minv[i] = mi;
            }
        }

        // Lane picks row M = (lane & 7) + 8*h via a cndmask tree
        const int j = lane & 7;
        const int M = j + 8 * h;
        const float val =
            (j & 4) ? ((j & 2) ? ((j & 1) ? minv[7] : minv[6])
                               : ((j & 1) ? minv[5] : minv[4]))
                    : ((j & 2) ? ((j & 1) ? minv[3] : minv[2])
                               : ((j & 1) ? minv[1] : minv[0]));
        const float a2r  = __shfl(a2, M, 32);             // |p|^2 of row M
        const float dist = sqrtf(fmaxf(val + a2r, 0.f));
        const int   gn   = rt * 16 + M;
        if (((lane & 8) == 0) && (gn < N_PTS)) rowSum += dist;
    }

    // wave sum -> block sum -> dmin
    #pragma unroll
    for (int mask = 16; mask > 0; mask >>= 1)
        rowSum += __shfl_xor(rowSum, mask, 32);
    if (lane == 0) sRed[wave] = rowSum;
    __syncthreads();
    if (tid == 0) {
        float s = 0.f;
        #pragma unroll
        for (int i = 0; i < 8; ++i) s += sRed[i];
        dmin_out[blockIdx.x] = s / (float)N_PTS;
    }
}

// ---------------------------------------------------------------------------
// Kernel 2: single-block deterministic finalize: loss_reg, loss_r, loss_t, total
// ---------------------------------------------------------------------------
__global__ __launch_bounds__(256)
void finalize_kernel(const float* __restrict__ pred_r,
                     const float* __restrict__ pred_c,
                     const float* __restrict__ rot_anchors,
                     const float* __restrict__ diameters,
                     const float* __restrict__ pred_t,
                     const float* __restrict__ target_t,
                     const int*   __restrict__ choose,
                     const float* __restrict__ dmin,
                     float* __restrict__ out)
{
    __shared__ float sR[256], sG[256], sT[256];
    const int tid = threadIdx.x;

    float accR = 0.f, accReg = 0.f;
    if (tid < BS * NUM_ROT) {
        const int b = tid / NUM_ROT;
        const int r = tid % NUM_ROT;
        const float q0 = pred_r[tid * 4 + 0];
        const float q1 = pred_r[tid * 4 + 1];
        const float q2 = pred_r[tid * 4 + 2];
        const float q3 = pred_r[tid * 4 + 3];
        float mx = -3.0e38f, diag = 0.f;
        for (int a = 0; a < NUM_ROT; ++a) {
            const float d = q0 * rot_anchors[a * 4 + 0] + q1 * rot_anchors[a * 4 + 1] +
                            q2 * rot_anchors[a * 4 + 2] + q3 * rot_anchors[a * 4 + 3];
            mx = fmaxf(mx, d);
            if (a == r) diag = d;
        }
        const float gap = mx - diag;
        accReg = (gap > 0.001f) ? gap : 0.f;                 // /96 later
        const float c = pred_c[tid];
        accR = dmin[tid] / (diameters[b] * c) + logf(c);     // /NUM_ROT later
    }

    // translation smooth-L1 with gather from NHWC-flattened target_t
    float accT = 0.f;
    for (int idx = tid; idx < BS * N_SEL * 3; idx += 256) {
        const int b   = idx / (N_SEL * 3);
        const int rem = idx - b * (N_SEL * 3);
        const int s   = rem / 3;
        const int c   = rem - s * 3;
        const int flat = choose[b * N_SEL + s];
        const int b2 = flat / (IMG_H * IMG_W);
        const int pp = flat - b2 * (IMG_H * IMG_W);
        const int hh = pp / IMG_W;
        const int ww = pp - hh * IMG_W;
        const float tv = target_t[((b2 * 3 + c) * IMG_H + hh) * IMG_W + ww];
        const float pv = pred_t[idx];
        const float d = fabsf(pv - tv);
        accT += (d < 1.f) ? (0.5f * d * d) : (d - 0.5f);
    }

    sR[tid] = accR; sG[tid] = accReg; sT[tid] = accT;
    __syncthreads();
    for (int s = 128; s > 0; s >>= 1) {
        if (tid < s) {
            sR[tid] += sR[tid + s];
            sG[tid] += sG[tid + s];
            sT[tid] += sT[tid + s];
        }
        __syncthreads();
    }
    if (tid == 0) {
        const float loss_r   = sR[0] / (float)NUM_ROT;
        const float loss_reg = sG[0] / (float)(BS * NUM_ROT);
        const float loss_t   = sT[0] / (float)(BS * N_SEL * 3);
        out[0] = loss_r + 2.f * loss_reg + 5.f * loss_t;
        out[1] = loss_r;
        out[2] = loss_reg;
        out[3] = loss_t;
    }
}

extern "C" void kernel_launch(void* const* d_in, const int* in_sizes, int n_in,
                              void* d_out, int out_size, void* d_ws, size_t ws_size,
                              hipStream_t stream)
{
    const float* pred_t       = (const float*)d_in[0];
    const float* pred_r       = (const float*)d_in[1];
    const float* pred_c       = (const float*)d_in[2];
    const float* target_r     = (const float*)d_in[3];
    const float* target_t     = (const float*)d_in[4];
    const float* model_points = (const float*)d_in[5];
    const int*   choose       = (const int*)d_in[6];
    const float* diameters    = (const float*)d_in[7];
    const float* rot_anchors  = (const float*)d_in[8];

    float* dmin = (float*)d_ws;   // 96 floats of scratch

    rot_dmin_kernel<<<BS * NUM_ROT, 256, 0, stream>>>(pred_r, model_points, target_r, dmin);
    finalize_kernel<<<1, 256, 0, stream>>>(pred_r, pred_c, rot_anchors, diameters,
                                           pred_t, target_t, choose, dmin, (float*)d_out);
}